// SignalingModel_89060441850572
// MI455X (gfx1250) — compile-verified
//
#include <hip/hip_runtime.h>

typedef float v2f __attribute__((ext_vector_type(2)));
typedef float v8f __attribute__((ext_vector_type(8)));

#define NNODES 2048   // N
#define NSAMP  512    // B
#define MAXSTEPS 60
#define LEAK 0.01f

#define BM 64
#define BN 64
#define KC 16
#define LDS_STRIDE 80   // float2 per pair-row; 160 floats == 32-bank shift/row -> conflict-free halves

__device__ __forceinline__ float mml_act(float x) {
    // leak*x for x<0 ; x for 0<=x<0.5 ; 1 - 0.25/x for x>=0.5 (x>=0.5 -> safe divide)
    float hi = 1.0f - 0.25f / x;
    float r  = (x < 0.5f) ? x : hi;
    return (x < 0.0f) ? LEAK * x : r;
}

// Step 1: X_bias = X_full^T + bias ; X1 = mml(X_bias)   (X0 == 0)
__global__ void init_kernel(const float* __restrict__ Xfull,   // (B, N) row-major
                            const float* __restrict__ bias,    // (N)
                            float* __restrict__ Xbias,         // (N, B) row-major
                            float* __restrict__ Xcur) {        // (N, B)
    int id = blockIdx.x * blockDim.x + threadIdx.x;            // 0 .. N*B-1
    int n = id >> 9;            // /512
    int b = id & (NSAMP - 1);
    float v = Xfull[(size_t)b * NNODES + n] + bias[n];
    Xbias[id] = v;
    Xcur[id]  = mml_act(v);
}

// One recurrence step: Xout = mml(W @ Xin + Xbias), all (N,B) row-major, W (N,N) row-major.
// Double-buffered LDS pipeline: one barrier per K-chunk; LDS stores for chunk i+1 and
// global prefetch for chunk i+2 overlap the 16 WMMAs on chunk i.
__global__ __launch_bounds__(128)
void gemm_step(const float* __restrict__ W,
               const float* __restrict__ Xin,
               const float* __restrict__ Xbias,
               float* __restrict__ Xout) {
    // K-pair-packed tiles:
    //   Wp[buf][kp][m] = ( W[m][2kp], W[m][2kp+1] )   (W transposed into LDS)
    //   Xp[buf][kp][n] = ( X[2kp][n], X[2kp+1][n] )
    __shared__ float2 Wp[2][KC / 2][LDS_STRIDE];
    __shared__ float2 Xp[2][KC / 2][LDS_STRIDE];

    const int tid  = threadIdx.x;
    const int lane = tid & 31;
    const int wave = tid >> 5;        // 0..3
    const int wm   = wave & 1;        // wave M half (32 rows)
    const int wn   = wave >> 1;       // wave N half (32 cols)
    const int l15  = lane & 15;
    const int lhi  = lane >> 4;       // 0 for lanes 0-15, 1 for lanes 16-31

    const int m0 = blockIdx.y * BM;
    const int n0 = blockIdx.x * BN;

    // --- global load mapping (per K-chunk of 16) ---
    const int wr = tid >> 2;                 // 0..31
    const int wk = (tid & 3) << 2;           // 0,4,8,12
    const float* wptr0 = W + (size_t)(m0 + wr)      * NNODES + wk;
    const float* wptr1 = W + (size_t)(m0 + wr + 32) * NNODES + wk;
    const int xk0 = tid >> 4;                // 0..7
    const int xc  = (tid & 15) << 2;         // 0..60
    const float* xptr0 = Xin + (size_t)xk0       * NSAMP + n0 + xc;
    const float* xptr1 = Xin + (size_t)(xk0 + 8) * NSAMP + n0 + xc;

    v8f acc00 = {0.f,0.f,0.f,0.f,0.f,0.f,0.f,0.f};
    v8f acc01 = acc00, acc10 = acc00, acc11 = acc00;

    float4 wA, wB, xA, xB;

    auto GLOAD = [&](int kc) {
        wA = *(const float4*)(wptr0 + kc);
        wB = *(const float4*)(wptr1 + kc);
        xA = *(const float4*)(xptr0 + kc * NSAMP);
        xB = *(const float4*)(xptr1 + kc * NSAMP);
    };
    auto LSTORE = [&](int b) {
        Wp[b][(wk >> 1)    ][wr     ] = make_float2(wA.x, wA.y);
        Wp[b][(wk >> 1) + 1][wr     ] = make_float2(wA.z, wA.w);
        Wp[b][(wk >> 1)    ][wr + 32] = make_float2(wB.x, wB.y);
        Wp[b][(wk >> 1) + 1][wr + 32] = make_float2(wB.z, wB.w);
        float* xd = ((float*)&Xp[b][xk0 >> 1][xc]) + (xk0 & 1);
        xd[0] = xA.x; xd[2] = xA.y; xd[4] = xA.z; xd[6] = xA.w;
        float* xe = ((float*)&Xp[b][(xk0 >> 1) + 4][xc]) + (xk0 & 1);
        xe[0] = xB.x; xe[2] = xB.y; xe[4] = xB.z; xe[6] = xB.w;
    };

    // Prologue: chunk 0 into buffer 0, prefetch chunk 1 into registers.
    GLOAD(0);
    LSTORE(0);
    GLOAD(KC);
    __syncthreads();

    int buf = 0;
    for (int kc = 0; kc < NNODES; kc += KC) {
        // Stage chunk kc+KC into the idle buffer; prefetch chunk kc+2*KC.
        if (kc + KC < NNODES) {
            LSTORE(buf ^ 1);
            if (kc + 2 * KC < NNODES) GLOAD(kc + 2 * KC);
        }
        // 16 WMMAs on the current buffer.
        #pragma unroll
        for (int k0 = 0; k0 < KC; k0 += 4) {
            const int kp = (k0 >> 1) + lhi;   // pair-row for this lane half (K = k0 + 2*lhi)
            v2f a0 = *(const v2f*)&Wp[buf][kp][wm * 32 +      l15];
            v2f a1 = *(const v2f*)&Wp[buf][kp][wm * 32 + 16 + l15];
            v2f b0 = *(const v2f*)&Xp[buf][kp][wn * 32 +      l15];
            v2f b1 = *(const v2f*)&Xp[buf][kp][wn * 32 + 16 + l15];
            acc00 = __builtin_amdgcn_wmma_f32_16x16x4_f32(false, a0, false, b0, (short)0, acc00, false, false);
            acc01 = __builtin_amdgcn_wmma_f32_16x16x4_f32(false, a0, false, b1, (short)0, acc01, false, false);
            acc10 = __builtin_amdgcn_wmma_f32_16x16x4_f32(false, a1, false, b0, (short)0, acc10, false, false);
            acc11 = __builtin_amdgcn_wmma_f32_16x16x4_f32(false, a1, false, b1, (short)0, acc11, false, false);
        }
        __syncthreads();   // next chunk's LDS stores complete; current buffer free for reuse
        buf ^= 1;
    }

    // Epilogue: + X_bias, MML activation, store next state (N,B).
    // C/D layout: VGPR r -> M = r + 8*lhi, N = l15.
    const int mb = m0 + wm * 32;
    const int nb = n0 + wn * 32 + l15;
    #pragma unroll
    for (int mi = 0; mi < 2; ++mi) {
        #pragma unroll
        for (int ni = 0; ni < 2; ++ni) {
            const v8f& c = (mi == 0) ? (ni == 0 ? acc00 : acc01)
                                     : (ni == 0 ? acc10 : acc11);
            const int col = nb + ni * 16;
            #pragma unroll
            for (int r = 0; r < 8; ++r) {
                const int row = mb + mi * 16 + r + lhi * 8;
                const size_t o = (size_t)row * NSAMP + col;
                Xout[o] = mml_act(c[r] + Xbias[o]);
            }
        }
    }
}

// (N,B) -> (B,N) tiled transpose for the final output.
__global__ void transpose_kernel(const float* __restrict__ in, float* __restrict__ out) {
    __shared__ float t[32][33];
    const int n0 = blockIdx.x * 32;
    const int b0 = blockIdx.y * 32;
    const int tx = threadIdx.x, ty = threadIdx.y;
    #pragma unroll
    for (int j = 0; j < 32; j += 8)
        t[ty + j][tx] = in[(size_t)(n0 + ty + j) * NSAMP + b0 + tx];
    __syncthreads();
    #pragma unroll
    for (int j = 0; j < 32; j += 8)
        out[(size_t)(b0 + ty + j) * NNODES + n0 + tx] = t[tx][ty + j];
}

extern "C" void kernel_launch(void* const* d_in, const int* in_sizes, int n_in,
                              void* d_out, int out_size, void* d_ws, size_t ws_size,
                              hipStream_t stream) {
    (void)in_sizes; (void)n_in; (void)out_size; (void)ws_size;
    const float* Xfull = (const float*)d_in[0];   // (B, N)
    const float* W     = (const float*)d_in[1];   // (N, N)
    const float* bias  = (const float*)d_in[2];   // (N)
    float* out = (float*)d_out;

    float* Xbias = (float*)d_ws;                               // N*B floats
    float* Xping = (float*)d_out;                              // reuse d_out as scratch state
    float* Xpong = (float*)d_ws + (size_t)NNODES * NSAMP;      // N*B floats

    // Step 1: X_bias and X1 = mml(X_bias)
    init_kernel<<<(NNODES * NSAMP) / 256, 256, 0, stream>>>(Xfull, bias, Xbias, Xping);

    // Steps 2..60: 59 GEMM+activation steps (odd count -> final state lands in Xpong, not d_out)
    dim3 grid(NSAMP / BN, NNODES / BM);   // (8, 32)
    float* cur = Xping;
    float* nxt = Xpong;
    for (int s = 1; s < MAXSTEPS; ++s) {
        gemm_step<<<grid, 128, 0, stream>>>(W, cur, Xbias, nxt);
        float* tmp = cur; cur = nxt; nxt = tmp;
    }

    // Output: Xf^T  (cur == Xpong, in workspace)
    transpose_kernel<<<dim3(NNODES / 32, NSAMP / 32), dim3(32, 8), 0, stream>>>(cur, out);
}